// KKTLoss_33122787787141
// MI455X (gfx1250) — compile-verified
//
#include <hip/hip_runtime.h>
#include <hip/hip_bf16.h>

// Problem sizes (match reference)
#define BB 32
#define MM 2048
#define NN 2048

typedef __attribute__((ext_vector_type(2))) float v2f;
typedef __attribute__((ext_vector_type(8))) float v8f;

// ---------------------------------------------------------------------------
// Kernel 0: zero workspace (Atlam accumulator [B*N] + 4 partial sums)
// ---------------------------------------------------------------------------
__global__ void kkt_zero_ws(float* ws, int n) {
    int i = blockIdx.x * blockDim.x + threadIdx.x;
    if (i < n) ws[i] = 0.0f;
}

// ---------------------------------------------------------------------------
// Kernel 1: single pass over A (one read of 512 MB = the roofline).
//  - A tiles staged global->LDS with async b128 copies (ASYNCcnt-tracked,
//    double buffered per wave) -> fully 128B-coalesced HBM traffic
//  - Atlam partials via v_wmma_f32_16x16x4_f32 (A-op = lam broadcast,
//    B-op = row-major A tile from LDS, 4 chained K-chunks per 16x16 tile)
//  - Ax via per-lane VALU fma on the same LDS-resident elements
//  - fused primal / dual / comp loss accumulation for finalized rows
// ---------------------------------------------------------------------------
__global__ void __launch_bounds__(256)
kkt_main(const float* __restrict__ A,
         const float* __restrict__ x,
         const float* __restrict__ lam,
         const float* __restrict__ b_pad,
         float* __restrict__ Atlam,   // [B*N] global accumulator
         float* __restrict__ sums)    // [0]=primal [1]=dual [2]=comp
{
    const int wg   = blockIdx.x;          // 0 .. B*16-1
    const int b    = wg >> 4;             // batch
    const int mblk = (wg & 15) * 128;     // 128-row block
    const int tid  = threadIdx.x;
    const int wave = tid >> 5;            // 0..7
    const int lane = tid & 31;
    const int half = lane >> 4;           // 0 | 1
    const int lc   = lane & 15;           // column within 16-wide tile
    const int m0   = mblk + wave * 16;    // this wave's 16-row strip

    __shared__ float zacc[NN];            // 8 KB column accumulator
    __shared__ float tbuf[8][2][256];     // 16 KB: per-wave double-buffered tile

    for (int i = tid; i < NN; i += 256) zacc[i] = 0.0f;
    __syncthreads();

    const float* Ab   = A   + (size_t)b * MM * NN;
    const float* xb   = x   + (size_t)b * NN;
    const float* lamb = lam + (size_t)b * MM;

    // strip-local lambda values (wave-uniform)
    float lam_s[16];
#pragma unroll
    for (int k = 0; k < 16; ++k) lam_s[k] = lamb[m0 + k];

    // Ax partial accumulators: 8 rows per lane
    // row(j) = m0 + 4*(j>>1) + (j&1) + 2*half
    float yacc[8];
#pragma unroll
    for (int j = 0; j < 8; ++j) yacc[j] = 0.0f;

    // async stage of one 16x16 tile: 2 x b128 per lane-set (1 KB total)
    const int srow = lane >> 2;           // 0..7
    const int sc4  = (lane & 3) * 4;      // 0,4,8,12

#define STAGE_TILE(nt, db)                                                     \
    {                                                                          \
        const float* g0 = Ab + (size_t)(m0 + srow) * NN + (nt) * 16 + sc4;     \
        const float* g1 = Ab + (size_t)(m0 + 8 + srow) * NN + (nt) * 16 + sc4; \
        uint32_t l0 = (uint32_t)(uintptr_t)&tbuf[wave][db][srow * 16 + sc4];   \
        uint32_t l1 = (uint32_t)(uintptr_t)&tbuf[wave][db][(srow + 8) * 16 + sc4]; \
        asm volatile("global_load_async_to_lds_b128 %0, %1, off"               \
                     :: "v"(l0), "v"((uint64_t)(uintptr_t)g0) : "memory");     \
        asm volatile("global_load_async_to_lds_b128 %0, %1, off"               \
                     :: "v"(l1), "v"((uint64_t)(uintptr_t)g1) : "memory");     \
    }

    STAGE_TILE(0, 0)

    for (int nt = 0; nt < NN / 16; ++nt) {
        const int n0 = nt * 16;
        const int db = nt & 1;

        if (nt + 1 < NN / 16) {
            STAGE_TILE(nt + 1, db ^ 1)
            // 2 copies in flight for next tile; current tile's copies done
            asm volatile("s_wait_asynccnt 0x2" ::: "memory");
        } else {
            asm volatile("s_wait_asynccnt 0x0" ::: "memory");
        }

        const float* tb = &tbuf[wave][db][0];
        const float xv = xb[n0 + lc];

        v8f z = {0.f, 0.f, 0.f, 0.f, 0.f, 0.f, 0.f, 0.f};
#pragma unroll
        for (int mc = 0; mc < 4; ++mc) {
            // B operand (f32 4x16) from LDS: v0 = rows K={0|2}, v1 = K={1|3}
            const float a0 = tb[(4 * mc + 2 * half) * 16 + lc];
            const float a1 = tb[(4 * mc + 2 * half + 1) * 16 + lc];
            v2f Bop; Bop.x = a0; Bop.y = a1;

            // A operand (f32 16x4): lam broadcast over rows
            v2f Aop;
            Aop.x = lam_s[4 * mc + 2 * half];
            Aop.y = lam_s[4 * mc + 2 * half + 1];

            // D = A x B + C  (accumulate over the 4 K-chunks of this tile)
            z = __builtin_amdgcn_wmma_f32_16x16x4_f32(
                    false, Aop, false, Bop, (short)0, z, false, false);

            // Ax partials from the same elements
            yacc[2 * mc + 0] = fmaf(a0, xv, yacc[2 * mc + 0]);
            yacc[2 * mc + 1] = fmaf(a1, xv, yacc[2 * mc + 1]);
        }

        // z replicated across all D rows; element 0 of both lane-halves
        // holds z_partial[n0 + lc]. Add once per column into LDS.
        if (half == 0) atomicAdd(&zacc[n0 + lc], z[0]);
    }
#undef STAGE_TILE

    // reduce Ax partials across the 16 lanes of each half
#pragma unroll
    for (int j = 0; j < 8; ++j) {
#pragma unroll
        for (int s = 1; s < 16; s <<= 1)
            yacc[j] += __shfl_xor(yacc[j], s, 32);
    }

    // finalized rows -> fused loss contributions (one lane per half)
    if (lc == 0) {
        float s_primal = 0.f, s_dual = 0.f, s_comp = 0.f;
#pragma unroll
        for (int j = 0; j < 8; ++j) {
            const int m = m0 + 4 * (j >> 1) + (j & 1) + 2 * half;
            const float axmb = yacc[j] - b_pad[(size_t)b * MM + m];
            const float l    = lamb[m];
            const float rp = fmaxf(axmb, 0.f);
            s_primal += rp * rp;
            const float rd = fmaxf(-l, 0.f);
            s_dual += rd * rd;
            const float cm = l * axmb;
            s_comp += cm * cm;
        }
        atomicAdd(&sums[0], s_primal);
        atomicAdd(&sums[1], s_dual);
        atomicAdd(&sums[2], s_comp);
    }

    __syncthreads();
    // flush workgroup's Atlam contribution (one atomic per column per WG)
    for (int i = tid; i < NN; i += 256)
        atomicAdd(&Atlam[(size_t)b * NN + i], zacc[i]);
}

// ---------------------------------------------------------------------------
// Kernel 2: stationarity loss  sum((Atlam + c)^2)
// ---------------------------------------------------------------------------
__global__ void __launch_bounds__(256)
kkt_stat(const float* __restrict__ Atlam,
         const float* __restrict__ c_pad,
         float* __restrict__ sums)
{
    const int i = blockIdx.x * 256 + threadIdx.x;
    const float v = Atlam[i] + c_pad[i];
    float sq = v * v;
#pragma unroll
    for (int s = 1; s < 32; s <<= 1) sq += __shfl_xor(sq, s, 32);

    __shared__ float wsum[8];
    const int lane = threadIdx.x & 31, w = threadIdx.x >> 5;
    if (lane == 0) wsum[w] = sq;
    __syncthreads();
    if (threadIdx.x == 0) {
        float t = 0.f;
#pragma unroll
        for (int j = 0; j < 8; ++j) t += wsum[j];
        atomicAdd(&sums[3], t);
    }
}

// ---------------------------------------------------------------------------
// Kernel 3: combine weighted means into the scalar output
// ---------------------------------------------------------------------------
__global__ void kkt_final(const float* __restrict__ sums,
                          float* __restrict__ out)
{
    if (threadIdx.x == 0 && blockIdx.x == 0) {
        const float invBM = 1.0f / (float)(BB * MM);
        const float invBN = 1.0f / (float)(BB * NN);
        out[0] = 0.1f * sums[0] * invBM   // primal
               + 0.1f * sums[1] * invBM   // dual
               + 0.6f * sums[3] * invBN   // stationarity
               + 0.2f * sums[2] * invBM;  // complementarity
    }
}

// ---------------------------------------------------------------------------
extern "C" void kernel_launch(void* const* d_in, const int* in_sizes, int n_in,
                              void* d_out, int out_size, void* d_ws, size_t ws_size,
                              hipStream_t stream)
{
    const float* x     = (const float*)d_in[0];   // x_hat   [B*N]
    const float* lam   = (const float*)d_in[1];   // lam_hat [B*M]
    const float* A     = (const float*)d_in[2];   // A       [B,M,N]
    const float* b_pad = (const float*)d_in[3];   // [B,M]
    const float* c_pad = (const float*)d_in[4];   // [B,N]
    (void)in_sizes; (void)n_in; (void)out_size; (void)ws_size;

    float* ws    = (float*)d_ws;
    float* Atlam = ws;                   // B*N floats
    float* sums  = ws + (size_t)BB * NN; // 4 floats

    const int nzero = BB * NN + 4;
    kkt_zero_ws<<<(nzero + 255) / 256, 256, 0, stream>>>(ws, nzero);

    kkt_main<<<BB * (MM / 128), 256, 0, stream>>>(A, x, lam, b_pad, Atlam, sums);

    kkt_stat<<<(BB * NN) / 256, 256, 0, stream>>>(Atlam, c_pad, sums);

    kkt_final<<<1, 64, 0, stream>>>(sums, (float*)d_out);
}